// Spconv3x1LeakyReLUBatchNorm_45861660786970
// MI455X (gfx1250) — compile-verified
//
#include <hip/hip_runtime.h>

typedef __attribute__((ext_vector_type(2))) float v2f;
typedef __attribute__((ext_vector_type(8))) float v8f;

#define BATCH 2
#define S0 480
#define S1 360
#define S2 32
#define NPTS 400000
#define CCH 32
#define LOOKUP_ELEMS (BATCH * S0 * S1 * S2) /* 11,059,200 */
#define EPSV 1e-5f
#define SLOPE 0.01f
#define TILES ((NPTS + 15) / 16)              /* 25,000 */
#define WPB 8                                 /* waves per block */
#define CONV_BLOCKS ((TILES + WPB - 1) / WPB) /* 3,125 */

/* ws layout (bytes) */
#define WS_LOOKUP_OFF 0
#define WS_PARTIAL_OFF (LOOKUP_ELEMS * 4)                    /* 44,236,800 */
#define WS_SCALE_OFF (WS_PARTIAL_OFF + CONV_BLOCKS * 64 * 4) /* 45,036,800 */
#define WS_ZERO_OFF (WS_SCALE_OFF + 256)                     /* 45,037,056 (16B aligned) */

__global__ __launch_bounds__(256) void k_init_lookup(int* __restrict__ lookup,
                                                     float* __restrict__ zerorow) {
    int i = blockIdx.x * blockDim.x + threadIdx.x;
    if (i < LOOKUP_ELEMS) lookup[i] = -1;
    if (i < CCH) zerorow[i] = 0.0f;
}

__global__ __launch_bounds__(256) void k_scatter(const int* __restrict__ coords,
                                                 int* __restrict__ lookup) {
    int i = blockIdx.x * blockDim.x + threadIdx.x;
    if (i < NPTS) {
        int b  = coords[4 * i + 0];
        int x0 = coords[4 * i + 1];
        int x1 = coords[4 * i + 2];
        int x2 = coords[4 * i + 3];
        lookup[((b * S0 + x0) * S1 + x1) * S2 + x2] = i;
    }
}

__global__ __launch_bounds__(256) void k_conv(const float* __restrict__ feat,
                                              const int* __restrict__ coords,
                                              const float* __restrict__ weight,
                                              const int* __restrict__ lookup,
                                              const float* __restrict__ zerorow,
                                              float* __restrict__ out,
                                              float* __restrict__ partial) {
    /* W swizzled for direct ds_load_b64 B-fragments:
       idx = (((k*2 + hi)*2 + t)*8 + c)*32 + lo*2 + e
       value = weight[(k*32 + 4c + 2hi + e)*32 + 16t + lo]            */
    __shared__ float Wswz[9 * 1024]; /* 36 KB */
    __shared__ float redS[WPB][CCH];
    __shared__ float redQ[WPB][CCH];

    for (int j = threadIdx.x; j < 9 * 1024; j += 256) {
        const int e  = j & 1;
        const int lj = (j >> 1) & 15;
        const int c  = (j >> 5) & 7;
        const int t  = (j >> 8) & 1;
        const int h  = (j >> 9) & 1;
        const int k  = j >> 10;
        Wswz[j] = weight[(k * 32 + 4 * c + 2 * h + e) * 32 + 16 * t + lj];
    }
    __syncthreads();

    const int wave = threadIdx.x >> 5;
    const int lane = threadIdx.x & 31;
    const int lo = lane & 15;
    const int hi = lane >> 4;
    const int tile = blockIdx.x * WPB + wave;

    /* this lane owns A-matrix row M = lo (point tile*16+lo) */
    const int row = tile * 16 + lo;
    const bool rowok = (row < NPTS);
    const int4 cc = *(const int4*)(coords + 4 * (rowok ? row : 0));
    const int cb = cc.x, cx0 = cc.y, cx1 = cc.z, cx2 = cc.w;

    /* per-k LDS base for this lane's B fragments (t=0; t=1 at +1024B) */
    const float* wb0 = Wswz + hi * 512 + lo * 2;

    v8f acc0 = {}; /* D tile, cout 0..15  */
    v8f acc1 = {}; /* D tile, cout 16..31 */

    for (int k = 0; k < 9; ++k) {
        const int d0 = k / 3 - 1;
        const int d2 = k % 3 - 1;
        const int n0 = cx0 + d0;
        const int n2 = cx2 + d2;
        const bool inb = rowok & (n0 >= 0) & (n0 < S0) & (n2 >= 0) & (n2 < S2);
        const int n0c = min(max(n0, 0), S0 - 1);
        const int n2c = min(max(n2, 0), S2 - 1);
        const int nbr = lookup[((cb * S0 + n0c) * S1 + cx1) * S2 + n2c];
        /* branchless: invalid lanes read the zero row */
        const float* frow = (inb && nbr >= 0) ? (feat + (size_t)nbr * CCH) : zerorow;
        const float* frowh = frow + 2 * hi;
        const float* wb = wb0 + k * 1024;

#pragma unroll
        for (int c = 0; c < 8; ++c) {
            /* A (16x4 f32): lane holds M=lo, K = 4c+2hi+{0,1} — one b64 load */
            v2f a = *(const v2f*)(frowh + 4 * c);
            /* B (4x16 f32): contiguous pre-swizzled pairs — one ds_load_b64 each */
            v2f b0 = *(const v2f*)(wb + c * 32);
            v2f b1 = *(const v2f*)(wb + 256 + c * 32);
            acc0 = __builtin_amdgcn_wmma_f32_16x16x4_f32(
                false, a, false, b0, (short)0, acc0, false, false);
            acc1 = __builtin_amdgcn_wmma_f32_16x16x4_f32(
                false, a, false, b1, (short)0, acc1, false, false);
        }
    }

    /* LeakyReLU + per-lane channel partial sums (channels lo and 16+lo) */
    float s0 = 0.f, q0 = 0.f, s1 = 0.f, q1 = 0.f;
#pragma unroll
    for (int v = 0; v < 8; ++v) {
        float x = acc0[v]; x = (x >= 0.f) ? x : SLOPE * x; acc0[v] = x;
        s0 += x; q0 += x * x;
        float y = acc1[v]; y = (y >= 0.f) ? y : SLOPE * y; acc1[v] = y;
        s1 += y; q1 += y * y;
    }

    /* store: D lane holds (M = v + 8*hi, N = lo) */
#pragma unroll
    for (int v = 0; v < 8; ++v) {
        const int r = tile * 16 + v + 8 * hi;
        if (r < NPTS) {
            out[r * CCH + lo] = acc0[v];
            out[r * CCH + 16 + lo] = acc1[v];
        }
    }

    /* deterministic wave reduction: combine the two half-waves */
    s0 += __shfl_xor(s0, 16, 32);
    q0 += __shfl_xor(q0, 16, 32);
    s1 += __shfl_xor(s1, 16, 32);
    q1 += __shfl_xor(q1, 16, 32);
    if (hi == 0) {
        redS[wave][lo] = s0; redS[wave][16 + lo] = s1;
        redQ[wave][lo] = q0; redQ[wave][16 + lo] = q1;
    }
    __syncthreads();

    /* per-block partials: [block][0..31]=sum, [block][32..63]=sumsq */
    if (threadIdx.x < 64) {
        const int ch = threadIdx.x & 31;
        const bool isQ = threadIdx.x >= 32;
        float acc = 0.f;
        for (int w = 0; w < WPB; ++w)
            acc += isQ ? redQ[w][ch] : redS[w][ch];
        partial[blockIdx.x * 64 + threadIdx.x] = acc;
    }
}

__global__ void k_finalize(const float* __restrict__ partial,
                           const float* __restrict__ gamma,
                           const float* __restrict__ beta,
                           float* __restrict__ scaleshift) {
    const int ch = threadIdx.x;
    if (ch < CCH) {
        double S = 0.0, Q = 0.0;
        for (int b = 0; b < CONV_BLOCKS; ++b) {
            S += (double)partial[b * 64 + ch];
            Q += (double)partial[b * 64 + 32 + ch];
        }
        const double invN = 1.0 / (double)NPTS;
        const float mean = (float)(S * invN);
        const float var = (float)(Q * invN) - mean * mean;
        const float sc = gamma[ch] * rsqrtf(var + EPSV);
        scaleshift[ch] = sc;
        scaleshift[CCH + ch] = beta[ch] - mean * sc;
    }
}

__global__ __launch_bounds__(256) void k_norm(float* __restrict__ out,
                                              const float* __restrict__ scaleshift) {
    int i = blockIdx.x * blockDim.x + threadIdx.x;
    if (i < NPTS * CCH) {
        const int ch = i & (CCH - 1);
        out[i] = out[i] * scaleshift[ch] + scaleshift[CCH + ch];
    }
}

extern "C" void kernel_launch(void* const* d_in, const int* in_sizes, int n_in,
                              void* d_out, int out_size, void* d_ws, size_t ws_size,
                              hipStream_t stream) {
    const float* feat = (const float*)d_in[0];
    const int* coords = (const int*)d_in[1];
    const float* weight = (const float*)d_in[2];
    const float* gamma = (const float*)d_in[3];
    const float* beta = (const float*)d_in[4];
    float* out = (float*)d_out;
    char* ws = (char*)d_ws;

    int* lookup = (int*)(ws + WS_LOOKUP_OFF);
    float* partial = (float*)(ws + WS_PARTIAL_OFF);
    float* scaleshift = (float*)(ws + WS_SCALE_OFF);
    float* zerorow = (float*)(ws + WS_ZERO_OFF);

    k_init_lookup<<<LOOKUP_ELEMS / 256, 256, 0, stream>>>(lookup, zerorow);
    k_scatter<<<(NPTS + 255) / 256, 256, 0, stream>>>(coords, lookup);
    k_conv<<<CONV_BLOCKS, 256, 0, stream>>>(feat, coords, weight, lookup, zerorow, out, partial);
    k_finalize<<<1, 32, 0, stream>>>(partial, gamma, beta, scaleshift);
    k_norm<<<(NPTS * CCH + 255) / 256, 256, 0, stream>>>(out, scaleshift);
}